// ChebNet_87222195847851
// MI455X (gfx1250) — compile-verified
//
#include <hip/hip_runtime.h>
#include <hip/hip_bf16.h>
#include <math.h>

typedef __attribute__((ext_vector_type(2))) float v2f;
typedef __attribute__((ext_vector_type(8))) float v8f;

#define IN_C  128
#define HID_C 64
#define OUT_C 32

// ---------------- small elementwise kernels ----------------

__global__ void fill_zero_u32(unsigned* p, int n) {
  int i = blockIdx.x * blockDim.x + threadIdx.x;
  if (i < n) p[i] = 0u;
}

__global__ void fill_zero_f(float* p, long long n) {
  long long i = (long long)blockIdx.x * blockDim.x + threadIdx.x;
  if (i < n) p[i] = 0.0f;
}

// dst[i] = -src[i]; safe for dst == src (each thread touches one index)
__global__ void neg_copy(float* dst, const float* src, long long n) {
  long long i = (long long)blockIdx.x * blockDim.x + threadIdx.x;
  if (i < n) dst[i] = -src[i];
}

__global__ void bias_init(float* __restrict__ o, const float* __restrict__ b,
                          int Cmask, long long n) {
  long long i = (long long)blockIdx.x * blockDim.x + threadIdx.x;
  if (i < n) o[i] = b[(int)(i & Cmask)];
}

__global__ void relu_ip(float* p, long long n) {
  long long i = (long long)blockIdx.x * blockDim.x + threadIdx.x;
  if (i < n) p[i] = fmaxf(p[i], 0.0f);
}

// ---------------- graph preprocessing ----------------

__global__ void deg_count(const int* __restrict__ rowi, unsigned* __restrict__ deg, int E) {
  int e = blockIdx.x * blockDim.x + threadIdx.x;
  if (e < E) atomicAdd(&deg[rowi[e]], 1u);
}

__global__ void compute_dis(const unsigned* __restrict__ deg, float* __restrict__ dis, int n) {
  int i = blockIdx.x * blockDim.x + threadIdx.x;
  if (i < n) {
    unsigned d = deg[i];
    dis[i] = (d > 0u) ? rsqrtf((float)d) : 0.0f;
  }
}

__global__ void compute_w(const int* __restrict__ rowi, const int* __restrict__ coli,
                          const float* __restrict__ dis, float* __restrict__ w, int E) {
  int e = blockIdx.x * blockDim.x + threadIdx.x;
  if (e < E) w[e] = -dis[rowi[e]] * dis[coli[e]];
}

// ---------------- SpMM scatter: o[row[e]][:] += scale*w[e]*h[col[e]][:] -----
// One thread handles one (edge, 4-channel chunk). Consecutive threads cover one
// edge's channels -> coalesced float4 gather, 4 f32 atomics to L2.

__global__ void spmm_scatter(float* __restrict__ o, const float* __restrict__ h,
                             const int* __restrict__ rowi, const int* __restrict__ coli,
                             const float* __restrict__ w, float scale,
                             int E, int C, int lgC4) {
  long long gid = (long long)blockIdx.x * blockDim.x + threadIdx.x;
  int C4 = C >> 2;
  long long total = (long long)E * C4;
  if (gid >= total) return;
  int e = (int)(gid >> lgC4);
  int chunk = (int)(gid & (C4 - 1));
  float we = w[e] * scale;
  int r = rowi[e], c = coli[e];
  float4 hv = ((const float4*)(h + (size_t)c * C))[chunk];
  float* dst = o + (size_t)r * C + (chunk << 2);
  atomicAdd(dst + 0, we * hv.x);
  atomicAdd(dst + 1, we * hv.y);
  atomicAdd(dst + 2, we * hv.z);
  atomicAdd(dst + 3, we * hv.w);
}

// ---------------- dense GEMM-accumulate via V_WMMA_F32_16X16X4_F32 ----------
// O[M,C] += T[M,K] @ W[K,C].  One wave per 16x16 tile, 4 waves / block.
// A tile (16x4 f32):  lanes 0-15 -> K = {k,k+1}; lanes 16-31 -> K = {k+2,k+3}
// B tile (4x16 f32):  same K split across lane halves, N = lane&15
// C/D (16x16 f32, 8 VGPRs): VGPR r -> M = r + 8*(lane>>4), N = lane&15

__global__ void __launch_bounds__(128)
wmma_gemm_acc(const float* __restrict__ T, const float* __restrict__ W,
              float* __restrict__ O, int M, int K, int C) {
  int lane = threadIdx.x & 31;
  int wave = threadIdx.x >> 5;
  int tile = blockIdx.x * 4 + wave;
  int tilesN = C >> 4;
  int totTiles = ((M + 15) >> 4) * tilesN;
  if (tile >= totTiles) return;               // wave-uniform: EXEC stays all-1s
  int tm = tile / tilesN;
  int tn = tile - tm * tilesN;

  int hi = lane >> 4;                         // 0: lanes 0-15, 1: lanes 16-31
  int lo = lane & 15;
  int arow = tm * 16 + lo;
  if (arow >= M) arow = M - 1;                // clamp loads for partial tiles
  const float* tp = T + (size_t)arow * K + (hi << 1);
  const float* wp = W + (size_t)(hi << 1) * C + tn * 16 + lo;

  v8f acc = {};
  for (int k = 0; k < K; k += 4) {
    v2f a, b;
    a.x = tp[k];
    a.y = tp[k + 1];
    b.x = wp[(size_t)k * C];
    b.y = wp[(size_t)(k + 1) * C];
    acc = __builtin_amdgcn_wmma_f32_16x16x4_f32(
        /*neg_a=*/false, a, /*neg_b=*/false, b,
        /*c_mod=*/(short)0, acc, /*reuse_a=*/false, /*reuse_b=*/false);
  }

  int orow = tm * 16 + (hi << 3);
  float* op = O + (size_t)orow * C + tn * 16 + lo;
#pragma unroll
  for (int r = 0; r < 8; ++r) {
    if (orow + r < M) op[(size_t)r * C] += acc[r];
  }
}

// ---------------- row-wise log_softmax, 32 cols == 32 lanes (wave32) --------

__global__ void log_softmax32(float* __restrict__ o, int M) {
  int row = blockIdx.x * (blockDim.x >> 5) + (threadIdx.x >> 5);
  int lane = threadIdx.x & 31;
  if (row >= M) return;
  float v = o[(size_t)row * 32 + lane];
  float m = v;
#pragma unroll
  for (int off = 16; off > 0; off >>= 1) m = fmaxf(m, __shfl_xor(m, off, 32));
  float e = expf(v - m);
  float s = e;
#pragma unroll
  for (int off = 16; off > 0; off >>= 1) s += __shfl_xor(s, off, 32);
  o[(size_t)row * 32 + lane] = v - m - logf(s);
}

// ---------------- host orchestration ----------------

extern "C" void kernel_launch(void* const* d_in, const int* in_sizes, int n_in,
                              void* d_out, int out_size, void* d_ws, size_t ws_size,
                              hipStream_t stream) {
  const float* x  = (const float*)d_in[0];   // [N, 128]
  const int*   ei = (const int*)d_in[1];     // [2, E]
  const float* W1 = (const float*)d_in[2];   // [4, 128, 64]
  const float* b1 = (const float*)d_in[3];   // [64]
  const float* W2 = (const float*)d_in[4];   // [4, 64, 32]
  const float* b2 = (const float*)d_in[5];   // [32]
  float* out = (float*)d_out;                // [N, 32]

  const int N = in_sizes[0] / IN_C;
  const int E = in_sizes[1] / 2;
  const int* rowi = ei;
  const int* coli = ei + E;

  // workspace carve-up
  float*    A    = (float*)d_ws;                       // [N,128] ping
  float*    B    = A + (size_t)N * IN_C;               // [N,128] pong
  float*    H    = B + (size_t)N * IN_C;               // [N,64]  layer1 out
  float*    dis  = H + (size_t)N * HID_C;              // [N]
  unsigned* deg  = (unsigned*)(dis + N);               // [N]
  float*    wbuf = (float*)(deg + N);                  // [E]

  const int TB = 256;
  auto nb = [](long long n, int tb) { return (unsigned)((n + tb - 1) / tb); };
  auto gemm = [&](const float* T, const float* Wk, float* O, int K, int C) {
    int tiles = ((N + 15) / 16) * (C / 16);
    wmma_gemm_acc<<<(tiles + 3) / 4, 128, 0, stream>>>(T, Wk, O, N, K, C);
  };

  long long n128 = (long long)N * IN_C;
  long long nH   = (long long)N * HID_C;
  long long nO   = (long long)N * OUT_C;

  // --- degree, dis, per-edge weights (computed once, reused by all 6 props)
  fill_zero_u32<<<nb(N, TB), TB, 0, stream>>>(deg, N);
  deg_count<<<nb(E, TB), TB, 0, stream>>>(rowi, deg, E);
  compute_dis<<<nb(N, TB), TB, 0, stream>>>(deg, dis, N);
  compute_w<<<nb(E, TB), TB, 0, stream>>>(rowi, coli, dis, wbuf, E);

  // ---------------- layer 1: Cin=128 -> 64 ----------------
  // T1 = prop(x) -> A
  fill_zero_f<<<nb(n128, TB), TB, 0, stream>>>(A, n128);
  spmm_scatter<<<nb((long long)E * (IN_C / 4), TB), TB, 0, stream>>>(
      A, x, rowi, coli, wbuf, 1.0f, E, IN_C, 5);
  // H = b1 + T0@W1[0] + T1@W1[1]
  bias_init<<<nb(nH, TB), TB, 0, stream>>>(H, b1, HID_C - 1, nH);
  gemm(x, W1 + 0 * IN_C * HID_C, H, IN_C, HID_C);
  gemm(A, W1 + 1 * IN_C * HID_C, H, IN_C, HID_C);
  // T2 = 2*prop(T1) - T0 -> B   (init B = -x, scatter 2w*A on top)
  neg_copy<<<nb(n128, TB), TB, 0, stream>>>(B, x, n128);
  spmm_scatter<<<nb((long long)E * (IN_C / 4), TB), TB, 0, stream>>>(
      B, A, rowi, coli, wbuf, 2.0f, E, IN_C, 5);
  gemm(B, W1 + 2 * IN_C * HID_C, H, IN_C, HID_C);
  // T3 = 2*prop(T2) - T1 -> A   (in-place negate A, scatter 2w*B)
  neg_copy<<<nb(n128, TB), TB, 0, stream>>>(A, A, n128);
  spmm_scatter<<<nb((long long)E * (IN_C / 4), TB), TB, 0, stream>>>(
      A, B, rowi, coli, wbuf, 2.0f, E, IN_C, 5);
  gemm(A, W1 + 3 * IN_C * HID_C, H, IN_C, HID_C);
  relu_ip<<<nb(nH, TB), TB, 0, stream>>>(H, nH);

  // ---------------- layer 2: Cin=64 -> 32 ----------------
  fill_zero_f<<<nb(nH, TB), TB, 0, stream>>>(A, nH);
  spmm_scatter<<<nb((long long)E * (HID_C / 4), TB), TB, 0, stream>>>(
      A, H, rowi, coli, wbuf, 1.0f, E, HID_C, 4);
  bias_init<<<nb(nO, TB), TB, 0, stream>>>(out, b2, OUT_C - 1, nO);
  gemm(H, W2 + 0 * HID_C * OUT_C, out, HID_C, OUT_C);
  gemm(A, W2 + 1 * HID_C * OUT_C, out, HID_C, OUT_C);
  neg_copy<<<nb(nH, TB), TB, 0, stream>>>(B, H, nH);
  spmm_scatter<<<nb((long long)E * (HID_C / 4), TB), TB, 0, stream>>>(
      B, A, rowi, coli, wbuf, 2.0f, E, HID_C, 4);
  gemm(B, W2 + 2 * HID_C * OUT_C, out, HID_C, OUT_C);
  neg_copy<<<nb(nH, TB), TB, 0, stream>>>(A, A, nH);
  spmm_scatter<<<nb((long long)E * (HID_C / 4), TB), TB, 0, stream>>>(
      A, B, rowi, coli, wbuf, 2.0f, E, HID_C, 4);
  gemm(A, W2 + 3 * HID_C * OUT_C, out, HID_C, OUT_C);

  // log_softmax over 32 columns, one wave per row
  int rpb = TB / 32;
  log_softmax32<<<(N + rpb - 1) / rpb, TB, 0, stream>>>(out, N);
}